// EfficientGlobalPointer_88931592831404
// MI455X (gfx1250) — compile-verified
//
#include <hip/hip_runtime.h>
#include <hip/hip_bf16.h>

// Problem constants (from reference): B=16, S=512, H=1024, E=16, D=64
#define Bc   16
#define Sc   512
#define Hc   1024
#define Dc   64          // head dim d
#define Ec   16
#define NP   128         // 2*D, proj width
#define MR   (Bc * Sc)   // 8192 total rows

typedef float v2f __attribute__((ext_vector_type(2)));
typedef float v8f __attribute__((ext_vector_type(8)));

// ---------------------------------------------------------------------------
// Stage 1: proj = X @ Wp + bp  (8192 x 128), fp32 WMMA 16x16x4.
// Epilogue (in LDS): de-interleave -> qw/kw + interleaved RoPE; biasq = proj@Wq + bq.
// ---------------------------------------------------------------------------
__global__ void __launch_bounds__(256)
egp_stage1(const float* __restrict__ X,    // (8192, 1024)
           const float* __restrict__ Wp,   // (1024, 128)
           const float* __restrict__ bp,   // (128)
           const float* __restrict__ Wq,   // (128, 32)
           const float* __restrict__ bq,   // (32)
           float* __restrict__ qw,         // (8192, 64)
           float* __restrict__ kw,         // (8192, 64)
           float* __restrict__ biasq)      // (8192, 32)
{
    __shared__ float smem[16384];          // 64 KB: lX[128][64] | lW[64][128], reused as proj[128][128]
    float* lX = smem;
    float* lW = smem + 8192;

    const int tid = threadIdx.x;
    const int wv  = tid >> 5;              // wave 0..7 (wave32)
    const int l   = tid & 31;
    const int lr  = l & 15;                // lane row/col within 16
    const int lh  = l >> 4;                // half-select (K offset / M offset)
    const int m0  = blockIdx.x * 128;

    v8f acc[8];
    #pragma unroll
    for (int t = 0; t < 8; ++t) acc[t] = (v8f){};

    for (int kc = 0; kc < Hc; kc += 64) {
        // cooperative load X chunk 128x64 (float4, coalesced)
        {
            const int c4 = (tid & 15) * 4;
            const int rb = tid >> 4;
            #pragma unroll
            for (int i = 0; i < 8; ++i) {
                const int row = rb + i * 16;
                *(float4*)(lX + row * 64 + c4) =
                    *(const float4*)(X + (size_t)(m0 + row) * Hc + kc + c4);
            }
        }
        // cooperative load Wp chunk 64x128
        {
            const int c4 = (tid & 31) * 4;
            const int rb = tid >> 5;
            #pragma unroll
            for (int i = 0; i < 8; ++i) {
                const int row = rb + i * 8;
                *(float4*)(lW + row * NP + c4) =
                    *(const float4*)(Wp + (size_t)(kc + row) * NP + c4);
            }
        }
        __syncthreads();

        const float* aRow = lX + (16 * wv + lr) * 64;
        for (int kk = 0; kk < 64; kk += 4) {
            // A 16x4: K = v + 2*lh; adjacent pair -> single ds_load_b64
            const v2f a = *(const v2f*)(aRow + kk + 2 * lh);
            #pragma unroll
            for (int t = 0; t < 8; ++t) {  // B 4x16: B[k][n] = Wp[k][16t+lr]
                v2f b;
                b.x = lW[(kk + 2 * lh) * NP + 16 * t + lr];
                b.y = lW[(kk + 2 * lh + 1) * NP + 16 * t + lr];
                acc[t] = __builtin_amdgcn_wmma_f32_16x16x4_f32(
                    false, a, false, b, (short)0, acc[t], false, false);
            }
        }
        __syncthreads();
    }

    // dump proj tile (+bp) into LDS:   proj[128][128]
    #pragma unroll
    for (int t = 0; t < 8; ++t)
        #pragma unroll
        for (int r = 0; r < 8; ++r) {
            const int ml = 16 * wv + r + 8 * lh;   // C/D layout: M = r + 8*lh
            const int nl = 16 * t + lr;            // N = lane&15 within tile
            smem[ml * 128 + nl] = acc[t][r] + bp[nl];
        }
    __syncthreads();

    // epilogue: 2 threads per row; part0 = RoPE q/k, part1 = biasq row
    const int row  = tid & 127;
    const int part = tid >> 7;
    const int mg   = m0 + row;
    const float* p = smem + row * 128;

    if (part == 0) {
        const float s = (float)(mg & (Sc - 1));    // position within sequence
        #pragma unroll 4
        for (int f = 0; f < 32; ++f) {
            // inv_freq[f] = 10000^(-f/32) ; ln(10000)/32 = 0.2878230902f
            const float invf = __expf(-(float)f * 0.2878230902f);
            float sn, cs;
            __sincosf(s * invf, &sn, &cs);
            const float q0 = p[4 * f], q1 = p[4 * f + 2];      // even channels -> q
            const float k0 = p[4 * f + 1], k1 = p[4 * f + 3];  // odd channels  -> k
            qw[(size_t)mg * Dc + 2 * f]     = q0 * cs - q1 * sn;
            qw[(size_t)mg * Dc + 2 * f + 1] = q1 * cs + q0 * sn;
            kw[(size_t)mg * Dc + 2 * f]     = k0 * cs - k1 * sn;
            kw[(size_t)mg * Dc + 2 * f + 1] = k1 * cs + k0 * sn;
        }
    } else {
        for (int c = 0; c < 32; ++c) {
            float sum = bq[c];
            #pragma unroll 8
            for (int j = 0; j < 128; ++j) sum += p[j] * Wq[j * 32 + c];
            biasq[(size_t)mg * 32 + c] = sum;
        }
    }
}

// ---------------------------------------------------------------------------
// Stage 2: per (b, 128x128 tile) compute qk^T once via WMMA, then stream the
// 16 e-planes (bias + pad mask + causal tril) with non-temporal stores.
// ---------------------------------------------------------------------------
__global__ void __launch_bounds__(256)
egp_stage2(const float* __restrict__ qw,     // (8192, 64)
           const float* __restrict__ kw,     // (8192, 64)
           const float* __restrict__ biasq,  // (8192, 32)
           const float* __restrict__ amask,  // (16, 512)
           float* __restrict__ out)          // (16, 16, 512, 512)
{
    __shared__ float smem[16384];            // lQ[128][64] | lK[128][64]; lQ reused for bias
    float* lQ = smem;
    float* lK = smem + 8192;

    const int tid = threadIdx.x;
    const int wv  = tid >> 5;
    const int l   = tid & 31;
    const int lr  = l & 15;
    const int lh  = l >> 4;
    const int n0  = blockIdx.x * 128;
    const int m0  = blockIdx.y * 128;
    const int b   = blockIdx.z;

    // load q/k tiles (K=64 fits in one pass)
    {
        const int c4 = (tid & 15) * 4;
        const int rb = tid >> 4;
        #pragma unroll
        for (int i = 0; i < 8; ++i) {
            const int row = rb + i * 16;
            *(float4*)(lQ + row * 64 + c4) =
                *(const float4*)(qw + ((size_t)b * Sc + m0 + row) * Dc + c4);
            *(float4*)(lK + row * 64 + c4) =
                *(const float4*)(kw + ((size_t)b * Sc + n0 + row) * Dc + c4);
        }
    }
    __syncthreads();

    v8f acc[8];
    #pragma unroll
    for (int t = 0; t < 8; ++t) acc[t] = (v8f){};

    const float* aRow = lQ + (16 * wv + lr) * 64;
    for (int kk = 0; kk < 64; kk += 4) {
        // adjacent pairs, 8B aligned -> ds_load_b64 directly into VGPR pairs
        const v2f a = *(const v2f*)(aRow + kk + 2 * lh);
        #pragma unroll
        for (int t = 0; t < 8; ++t) {
            const v2f bb = *(const v2f*)(lK + (16 * t + lr) * 64 + kk + 2 * lh);
            acc[t] = __builtin_amdgcn_wmma_f32_16x16x4_f32(
                false, a, false, bb, (short)0, acc[t], false, false);
        }
    }
    __syncthreads();

    // stage bias/mask slices in reused LDS (lQ region)
    float* bm = smem;            // [128][16]: 0.5 * biasq[m, 2e+1]
    float* bn = smem + 2048;     // [128][16]: 0.5 * biasq[n, 2e]
    float* mk = smem + 4096;     // [128]    : attention_mask[b, n]
    for (int i = tid; i < 128 * 16; i += 256) {
        const int r = i >> 4, e = i & 15;
        bm[i] = 0.5f * biasq[((size_t)b * Sc + m0 + r) * 32 + 2 * e + 1];
        bn[i] = 0.5f * biasq[((size_t)b * Sc + n0 + r) * 32 + 2 * e];
    }
    if (tid < 128) mk[tid] = amask[(size_t)b * Sc + n0 + tid];
    __syncthreads();

    // stream 16 e-planes: the bandwidth-bound part. Output (256 MiB) > L2
    // (192 MB) and is never re-read -> non-temporal stores keep L2 for q/k/bias.
    const float INV_SQRT_D = 0.125f;       // 1/sqrt(64)
    for (int e = 0; e < 16; ++e) {
        #pragma unroll
        for (int t = 0; t < 8; ++t) {
            const int nl  = 16 * t + lr;
            const int ng  = n0 + nl;
            const float pad = mk[nl];
            const float cb  = bn[nl * 16 + e];
            const float neg = (1.0f - pad) * 1.0e12f;
            #pragma unroll
            for (int r = 0; r < 8; ++r) {
                const int ml = 16 * wv + r + 8 * lh;
                const int mg = m0 + ml;
                float v = acc[t][r] * INV_SQRT_D + cb + bm[ml * 16 + e];
                v = v * pad - neg;
                if (ng < mg) v -= 1.0e12f;     // tril(., -1) causal mask
                __builtin_nontemporal_store(
                    v, out + (((size_t)b * Ec + e) * Sc + mg) * Sc + ng);
            }
        }
    }
}

// ---------------------------------------------------------------------------
extern "C" void kernel_launch(void* const* d_in, const int* in_sizes, int n_in,
                              void* d_out, int out_size, void* d_ws, size_t ws_size,
                              hipStream_t stream) {
    (void)in_sizes; (void)n_in; (void)out_size; (void)ws_size;
    const float* X  = (const float*)d_in[0];   // last_hidden_state
    const float* am = (const float*)d_in[1];   // attention_mask
    const float* Wp = (const float*)d_in[2];
    const float* bp = (const float*)d_in[3];
    const float* Wq = (const float*)d_in[4];
    const float* bq = (const float*)d_in[5];
    float* out = (float*)d_out;

    float* qwp   = (float*)d_ws;               // 8192*64 floats = 2 MB
    float* kwp   = qwp + (size_t)MR * Dc;      // 2 MB
    float* bqp   = kwp + (size_t)MR * Dc;      // 8192*32 floats = 1 MB

    egp_stage1<<<dim3(MR / 128), dim3(256), 0, stream>>>(X, Wp, bp, Wq, bq, qwp, kwp, bqp);
    egp_stage2<<<dim3(Sc / 128, Sc / 128, Bc), dim3(256), 0, stream>>>(qwp, kwp, bqp, am, out);
}